// Gaussian_Embedding_36661840838908
// MI455X (gfx1250) — compile-verified
//
#include <hip/hip_runtime.h>

typedef float v2f __attribute__((ext_vector_type(2)));
typedef float v8f __attribute__((ext_vector_type(8)));

// Fixed problem shape: x is (N=64, C=3, W=16, T=128, V=25, M=2) f32.
constexpr int PER_N    = 128 * 25 * 2;      // T*V*M = 6400
constexpr int STRIDE_W = PER_N;             // 6400
constexpr int STRIDE_C = 16 * PER_N;        // 102400
constexpr int STRIDE_N = 3 * STRIDE_C;      // 307200
constexpr int B_TOTAL  = 64 * PER_N;        // 409600 blocks

constexpr int WAVES_PER_WG    = 4;
constexpr int THREADS         = WAVES_PER_WG * 32;   // 128
constexpr int BLOCKS_PER_WAVE = 32;                  // one block per lane
constexpr int BLOCKS_PER_WG   = WAVES_PER_WG * BLOCKS_PER_WAVE; // 128
constexpr int NUM_WG          = B_TOTAL / BLOCKS_PER_WG;        // 3200 exact

constexpr int XSLICE = 49;        // 48 floats per block tile, +1 pad (49 coprime 64 banks)
constexpr int DROW   = 17;        // 16x16 D tile rows padded to 17
constexpr int DBATCH = 16 * DROW; // 272 floats per WMMA batch
constexpr int NBATCH = 7;         // ceil(32 blocks / 5 blocks-per-WMMA-batch)

__global__ __launch_bounds__(THREADS)
void gauss_embed_kernel(const float* __restrict__ x, float* __restrict__ out)
{
    __shared__ float lds_x[BLOCKS_PER_WG * XSLICE];              // raw 3x16 tiles
    __shared__ float lds_mean[BLOCKS_PER_WG * 3];                // per-channel means
    __shared__ float lds_D[WAVES_PER_WG * NBATCH * DBATCH];      // WMMA D tiles

    const int tid  = threadIdx.x;
    const int wave = tid >> 5;
    const int lane = tid & 31;
    const int jwg  = tid;                                  // block slot in WG
    const int b    = blockIdx.x * BLOCKS_PER_WG + jwg;     // global block id

    // ---------------- Phase 1: coalesced load + means + LDS staging ----------
    // xr[b][c][w] corresponds to flat f = c*16+w read at (c'=f%3, w'=f/3).
    const int n    = b / PER_N;
    const int rem  = b - n * PER_N;
    const int base = n * STRIDE_N + rem;

    float s0 = 0.f, s1 = 0.f, s2 = 0.f;
    float* slice = &lds_x[jwg * XSLICE];
#pragma unroll
    for (int f = 0; f < 48; ++f) {
        const int cp = f % 3, wp = f / 3;
        const float v = x[base + cp * STRIDE_C + wp * STRIDE_W];
        if (f < 16)      s0 += v;
        else if (f < 32) s1 += v;
        else             s2 += v;
        slice[f] = v;
    }
    const float m0 = s0 * (1.0f / 16.0f);
    const float m1 = s1 * (1.0f / 16.0f);
    const float m2 = s2 * (1.0f / 16.0f);
    lds_mean[jwg * 3 + 0] = m0;
    lds_mean[jwg * 3 + 1] = m1;
    lds_mean[jwg * 3 + 2] = m2;
    __syncthreads();

    // ---------------- Phase 2: batched covariances via V_WMMA_F32_16X16X4_F32
    // A 16x4 layout: lane r (0..15) holds K=0,1 ; lane r+16 holds K=2,3.
    // Rows 3j+c (j=0..4 blocks, c=0..2 channels) -> D[3j+c][3j+d] = cov (unnorm).
    // All LDS loads are unconditional (clamped address); validity applied as a
    // multiplicative mask so no divergent branches are generated and the WMMA
    // executes with EXEC all-ones.
    const int r    = lane & 15;
    const int half = lane >> 4;
    float* waveD = &lds_D[wave * NBATCH * DBATCH];
#pragma unroll
    for (int t = 0; t < NBATCH; ++t) {
        const int   jj    = 5 * t + r / 3;                 // block within wave
        const int   ch    = r - 3 * (r / 3);
        const bool  valid = (r < 15) && (jj < BLOCKS_PER_WAVE);
        const float vmask = valid ? 1.0f : 0.0f;
        const int   jcl   = (jj < BLOCKS_PER_WAVE) ? jj : (BLOCKS_PER_WAVE - 1);
        const int   jloc  = wave * BLOCKS_PER_WAVE + jcl;  // always a safe slot

        const float  mv = lds_mean[jloc * 3 + ch];
        const float* xs = &lds_x[jloc * XSLICE + ch * 16];

        // Issue all 8 row loads for this batch up front (single dscnt wait).
        float xv[8];
#pragma unroll
        for (int kt = 0; kt < 4; ++kt) {
            xv[2 * kt]     = xs[4 * kt + 2 * half];
            xv[2 * kt + 1] = xs[4 * kt + 2 * half + 1];
        }

        v8f acc = {0.f, 0.f, 0.f, 0.f, 0.f, 0.f, 0.f, 0.f};
#pragma unroll
        for (int kt = 0; kt < 4; ++kt) {
            v2f a;
            a.x = (xv[2 * kt]     - mv) * vmask;
            a.y = (xv[2 * kt + 1] - mv) * vmask;
            // cov = Xc * Xc^T : A/B layouts are transpose-symmetric -> reuse regs.
            acc = __builtin_amdgcn_wmma_f32_16x16x4_f32(
                      false, a, false, a, (short)0, acc, false, false);
        }
        float* Dt = &waveD[t * DBATCH];
#pragma unroll
        for (int vi = 0; vi < 8; ++vi) {
            // D layout: VGPR vi, lanes 0-15 -> M=vi ; lanes 16-31 -> M=vi+8.
            Dt[(vi + 8 * half) * DROW + r] = acc[vi];
        }
    }
    __syncthreads();

    // ---------------- Phase 3: per-lane finalize (lane owns its block) -------
    const int t3 = lane / 5;
    const int r0 = 3 * (lane - 5 * t3);
    const float* Dme = &waveD[t3 * DBATCH];
    const float inv_w1 = 1.0f / 15.0f;   // 1/(W-1)

    const float c00 = Dme[(r0 + 0) * DROW + (r0 + 0)] * inv_w1;
    const float c01 = Dme[(r0 + 0) * DROW + (r0 + 1)] * inv_w1;
    const float c02 = Dme[(r0 + 0) * DROW + (r0 + 2)] * inv_w1;
    const float c10 = Dme[(r0 + 1) * DROW + (r0 + 0)] * inv_w1;
    const float c11 = Dme[(r0 + 1) * DROW + (r0 + 1)] * inv_w1;
    const float c12 = Dme[(r0 + 1) * DROW + (r0 + 2)] * inv_w1;
    const float c20 = Dme[(r0 + 2) * DROW + (r0 + 0)] * inv_w1;
    const float c21 = Dme[(r0 + 2) * DROW + (r0 + 1)] * inv_w1;
    const float c22 = Dme[(r0 + 2) * DROW + (r0 + 2)] * inv_w1;

    const float tr    = c00 + c11 + c22;
    const float itr   = 1.0f / tr;
    const float ridge = tr * 0.001f;
    const float n00 = c00 * itr + ridge, n01 = c01 * itr,         n02 = c02 * itr;
    const float n10 = c10 * itr,         n11 = c11 * itr + ridge, n12 = c12 * itr;
    const float n20 = c20 * itr,         n21 = c21 * itr,         n22 = c22 * itr + ridge;

    const float det = n00 * (n11 * n22 - n12 * n21)
                    - n01 * (n10 * n22 - n12 * n20)
                    + n02 * (n10 * n21 - n11 * n20);
    const float dsp = rsqrtf(sqrtf(det));   // det^(-1/4), det > 0 (PD + ridge)

    float4* o = (float4*)(out + (size_t)b * 16);
    o[0] = make_float4(dsp * (n00 + m0 * m0), dsp * (n01 + m0 * m1),
                       dsp * (n02 + m0 * m2), dsp * m0);
    o[1] = make_float4(dsp * (n10 + m1 * m0), dsp * (n11 + m1 * m1),
                       dsp * (n12 + m1 * m2), dsp * m1);
    o[2] = make_float4(dsp * (n20 + m2 * m0), dsp * (n21 + m2 * m1),
                       dsp * (n22 + m2 * m2), dsp * m2);
    o[3] = make_float4(dsp * m0, dsp * m1, dsp * m2, dsp);
}

extern "C" void kernel_launch(void* const* d_in, const int* in_sizes, int n_in,
                              void* d_out, int out_size, void* d_ws, size_t ws_size,
                              hipStream_t stream) {
    (void)in_sizes; (void)n_in; (void)out_size; (void)d_ws; (void)ws_size;
    const float* x = (const float*)d_in[0];
    float* out = (float*)d_out;
    gauss_embed_kernel<<<NUM_WG, THREADS, 0, stream>>>(x, out);
}